// GraphSAGELayer_45681272160469
// MI455X (gfx1250) — compile-verified
//
#include <hip/hip_runtime.h>

typedef __attribute__((ext_vector_type(2))) float v2f;
typedef __attribute__((ext_vector_type(8))) float v8f;

#define D 128           // D_IN == D_OUT == 128
#define TWO_D 256

// ---------------------------------------------------------------------------
// Kernel 1: zero neighbor_sum [N*128] and count [N] in workspace
// ---------------------------------------------------------------------------
__global__ void sage_zero_kernel(float* ws, long long total) {
    long long i = (long long)blockIdx.x * blockDim.x + threadIdx.x;
    if (i < total) ws[i] = 0.0f;
}

// ---------------------------------------------------------------------------
// Kernel 2: edge scatter-add.  One wave32 per edge: float4 gather of x[col]
// (512B coalesced per wave), 4 float atomics per lane into neighbor_sum[row].
// ---------------------------------------------------------------------------
__global__ __launch_bounds__(256)
void sage_scatter_kernel(const float* __restrict__ x,
                         const long long* __restrict__ ei,   // [2, E] row-major
                         float* __restrict__ nsum,
                         float* __restrict__ cnt,
                         int E) {
    const int e    = (int)(((long long)blockIdx.x * blockDim.x + threadIdx.x) >> 5);
    const int lane = threadIdx.x & 31;
    if (e >= E) return;
    const long long r = ei[e];              // destination node
    const long long c = ei[(long long)E + e]; // source neighbor
    const float4 v = *(const float4*)&x[c * D + lane * 4];
    float* dst = &nsum[r * D + lane * 4];
    atomicAdd(dst + 0, v.x);
    atomicAdd(dst + 1, v.y);
    atomicAdd(dst + 2, v.z);
    atomicAdd(dst + 3, v.w);
    if (lane == 0) atomicAdd(&cnt[r], 1.0f);
}

// ---------------------------------------------------------------------------
// Kernel 3: neighbor_sum -> neighbor_mean (in place)
// ---------------------------------------------------------------------------
__global__ void sage_mean_kernel(float* __restrict__ nsum,
                                 const float* __restrict__ cnt,
                                 long long total) {
    long long i = (long long)blockIdx.x * blockDim.x + threadIdx.x;
    if (i < total) {
        const long long n = i >> 7;   // / 128
        nsum[i] = nsum[i] / (cnt[n] + 1e-8f);
    }
}

// ---------------------------------------------------------------------------
// Kernel 4: out = relu([x | mean] @ W^T + b) via V_WMMA_F32_16X16X4_F32.
// Block = one M-tile of 16 rows; 8 waves cover the 8 N-tiles (D_OUT=128).
// A tiles (x and mean, 16x128 each) staged in LDS with stride 132 floats:
// 132 % 64 == 4 banks  =>  the 16-row strided v2f reads of both lane halves
// land on 64 distinct banks (conflict-free ds_load_b64).
//
// ISA VGPR layouts used (wave32):
//   A (16x4 f32):  lane<16: M=lane,    VGPR0/1 = K0,K1
//                  lane>=16: M=lane-16, VGPR0/1 = K2,K3
//   B (4x16 f32):  mirrored: lane half selects K-pair, lane&15 = N column
//   C/D (16x16):   VGPR j: M = j + 8*(lane>>4), N = lane&15
// ---------------------------------------------------------------------------
#define LDS_STRIDE 132

__global__ __launch_bounds__(256)
void sage_gemm_kernel(const float* __restrict__ x,
                      const float* __restrict__ mean,
                      const float* __restrict__ W,     // [128, 256] row-major
                      const float* __restrict__ bias,  // [128]
                      float* __restrict__ out,         // [N, 128]
                      int N) {
    __shared__ float xT[16 * LDS_STRIDE];
    __shared__ float mT[16 * LDS_STRIDE];

    const int lane = threadIdx.x & 31;
    const int wv   = threadIdx.x >> 5;     // 0..7 -> N-tile
    const int half = lane >> 4;            // K-pair selector
    const int mrow = lane & 15;
    const int m0   = blockIdx.x * 16;
    const int n0   = wv * 16;

    // Cooperative stage of the 16x128 A tiles (coalesced: consecutive threads
    // load consecutive columns).
    for (int i = threadIdx.x; i < 16 * D; i += 256) {
        const int r = i >> 7;
        const int c = i & (D - 1);
        const int gr = m0 + r;
        const bool ok = gr < N;
        xT[r * LDS_STRIDE + c] = ok ? x[(long long)gr * D + c] : 0.0f;
        mT[r * LDS_STRIDE + c] = ok ? mean[(long long)gr * D + c] : 0.0f;
    }
    __syncthreads();

    v8f acc = {};

    // K = 0..127 : self features (first half of W columns)
    #pragma unroll 8
    for (int k = 0; k < D; k += 4) {
        const int kk = k + 2 * half;
        const v2f a = *(const v2f*)&xT[mrow * LDS_STRIDE + kk];
        const v2f b = *(const v2f*)&W[(n0 + mrow) * TWO_D + kk];
        acc = __builtin_amdgcn_wmma_f32_16x16x4_f32(
            false, a, false, b, (short)0, acc, false, false);
    }
    // K = 128..255 : neighbor mean (second half of W columns)
    #pragma unroll 8
    for (int k = 0; k < D; k += 4) {
        const int kk = k + 2 * half;
        const v2f a = *(const v2f*)&mT[mrow * LDS_STRIDE + kk];
        const v2f b = *(const v2f*)&W[(n0 + mrow) * TWO_D + D + kk];
        acc = __builtin_amdgcn_wmma_f32_16x16x4_f32(
            false, a, false, b, (short)0, acc, false, false);
    }

    // Epilogue: bias + ReLU, scatter per C/D layout.
    const float bb = bias[n0 + mrow];
    #pragma unroll
    for (int j = 0; j < 8; ++j) {
        const int row = m0 + j + 8 * half;
        if (row < N) {
            float v = acc[j] + bb;
            out[(long long)row * D + n0 + mrow] = v > 0.0f ? v : 0.0f;
        }
    }
}

// ---------------------------------------------------------------------------
extern "C" void kernel_launch(void* const* d_in, const int* in_sizes, int n_in,
                              void* d_out, int out_size, void* d_ws, size_t ws_size,
                              hipStream_t stream) {
    const float*     x    = (const float*)d_in[0];
    const long long* ei   = (const long long*)d_in[1];
    const float*     W    = (const float*)d_in[2];
    const float*     bias = (const float*)d_in[3];
    float*           out  = (float*)d_out;

    const int N = in_sizes[0] / D;
    const int E = in_sizes[1] / 2;

    float* nsum = (float*)d_ws;                 // N * 128 floats
    float* cnt  = nsum + (long long)N * D;      // N floats

    const long long zeroTotal = (long long)N * D + N;
    {
        const int blk = 256;
        const int grd = (int)((zeroTotal + blk - 1) / blk);
        sage_zero_kernel<<<grd, blk, 0, stream>>>((float*)d_ws, zeroTotal);
    }
    {
        const int blk = 256;                    // 8 waves = 8 edges / block
        const int grd = (E + 7) / 8;
        sage_scatter_kernel<<<grd, blk, 0, stream>>>(x, ei, nsum, cnt, E);
    }
    {
        const long long total = (long long)N * D;
        const int blk = 256;
        const int grd = (int)((total + blk - 1) / blk);
        sage_mean_kernel<<<grd, blk, 0, stream>>>(nsum, cnt, total);
    }
    {
        const int blk = 256;                    // 8 waves = 8 N-tiles
        const int grd = (N + 15) / 16;          // one block per M-tile
        sage_gemm_kernel<<<grd, blk, 0, stream>>>(x, nsum, W, bias, out, N);
    }
}